// E67HGatedCell_4887672783130
// MI455X (gfx1250) — compile-verified
//
#include <hip/hip_runtime.h>
#include <math.h>

// ---------------------------------------------------------------------------
// E67HGatedCell for MI455X (gfx1250, wave32)
//   Phase 1: fp32 GEMMs via V_WMMA_F32_16X16X4_F32, LDS-staged with
//            GLOBAL_LOAD_ASYNC_TO_LDS_B128 (ASYNCcnt), 2x2 register blocking
//   Phase 2: bandwidth-bound sequential scan (512 MB streamed @ 23.3 TB/s)
// ---------------------------------------------------------------------------

typedef float v2f __attribute__((ext_vector_type(2)));
typedef float v4f __attribute__((ext_vector_type(4)));
typedef float v8f __attribute__((ext_vector_type(8)));
typedef int   v4i __attribute__((ext_vector_type(4)));

#define T_STEPS 1024
#define BATCH   32
#define DIM     1024
#define MTOT    (T_STEPS * BATCH)   // 32768 rows of x (flattened [T*B, D])
#define BD      (BATCH * DIM)       // 32768 elements per timestep

// GEMM tiling: block = 256 threads = 8 wave32s arranged 2(M) x 4(N);
// each wave computes a 32x32 macro-tile (2x2 WMMA tiles, 4 x v8f accum).
#define BM 64
#define BN 128
#define KC 32
#define KP (KC + 4)   // padded LDS row stride: 144 B (16B aligned, 0-conflict)

#if defined(__has_builtin)
#if __has_builtin(__builtin_amdgcn_global_load_async_to_lds_b128)
#define USE_ASYNC_LDS 1
#endif
#endif
#ifndef USE_ASYNC_LDS
#define USE_ASYNC_LDS 0
#endif

// Builtin prototype: (v4i addrspace(1)* src, v4i addrspace(3)* dst,
//                     imm offset, imm cpol)
typedef __attribute__((address_space(1))) v4i* gv4_ptr;
typedef __attribute__((address_space(3))) v4i* lv4_ptr;

__device__ __forceinline__ void copy16_to_lds(const float* g, float* l) {
#if USE_ASYNC_LDS
    // global_load_async_to_lds_b128: memory -> LDS without VGPR round-trip
    __builtin_amdgcn_global_load_async_to_lds_b128(
        (gv4_ptr)g, (lv4_ptr)l, /*imm offset*/0, /*cpol*/0);
#else
    *(v4f*)l = *(const v4f*)g;
#endif
}

__device__ __forceinline__ void wait_lds_fill() {
#if USE_ASYNC_LDS
#if defined(__has_builtin) && __has_builtin(__builtin_amdgcn_s_wait_asynccnt)
    __builtin_amdgcn_s_wait_asynccnt(0);
#else
    asm volatile("s_wait_asynccnt 0x0" ::: "memory");
#endif
#endif
    __syncthreads();
}

__device__ __forceinline__ v8f wmma4(v2f a, v2f b, v8f c) {
    return __builtin_amdgcn_wmma_f32_16x16x4_f32(
        false, a, false, b, (short)0, c, false, false);
}

__device__ __forceinline__ void store_tile(
    v8f acc, int m0, int n0, int rsel, int hi8,
    float* __restrict__ Y, const float* __restrict__ bias, int applyTanh)
{
    const int col = n0 + rsel;
    const float bn = bias ? bias[col] : 0.0f;
#pragma unroll
    for (int j = 0; j < 8; ++j) {
        float vv = acc[j];
        if (applyTanh) vv = tanhf(vv + bn);
        Y[(size_t)(m0 + hi8 + j) * DIM + col] = vv;
    }
}

// Y[m,n] = sum_k X[m,k] * W[n,k]   (optionally tanh(acc + bias[n]))
__global__ __launch_bounds__(256) void gemm_xwt_kernel(
    const float* __restrict__ X, const float* __restrict__ W,
    float* __restrict__ Y, const float* __restrict__ bias, int applyTanh)
{
    __shared__ float Xs[BM][KP];   //  9216 B
    __shared__ float Ws[BN][KP];   // 18432 B

    const int tid  = threadIdx.x;
    const int lane = tid & 31;
    const int wave = tid >> 5;                 // 0..7
    const int wm   = wave >> 2;                // 0..1  (M)
    const int wn   = wave & 3;                 // 0..3  (N)

    // f32 WMMA fragment layout: lanes 0-15 -> row=lane,    K={k,k+1}
    //                           lanes 16-31 -> row=lane-16, K={k+2,k+3}
    const int rsel = lane & 15;
    const int koff = (lane >> 4) * 2;
    const int hi8  = (lane >> 4) << 3;         // C/D: upper lanes hold M+8

    const int mblk = blockIdx.x * BM;
    const int nblk = blockIdx.y * BN;

    v8f acc00 = {}, acc01 = {}, acc10 = {}, acc11 = {};

    for (int k0 = 0; k0 < DIM; k0 += KC) {
        // Stage X tile (64x32 = 512 float4, 2/thread) and
        //       W tile (128x32 = 1024 float4, 4/thread) into LDS.
        {
            int i = tid;
#pragma unroll
            for (int r = 0; r < 2; ++r, i += 256) {
                const int row = i >> 3, c4 = (i & 7) << 2;
                copy16_to_lds(X + (size_t)(mblk + row) * DIM + k0 + c4,
                              &Xs[row][c4]);
            }
            i = tid;
#pragma unroll
            for (int r = 0; r < 4; ++r, i += 256) {
                const int row = i >> 3, c4 = (i & 7) << 2;
                copy16_to_lds(W + (size_t)(nblk + row) * DIM + k0 + c4,
                              &Ws[row][c4]);
            }
        }
        wait_lds_fill();

        // 8 K-steps: 4 ds_load_b64 + 4 WMMAs each (1:1 load:wmma, all-bank LDS)
#pragma unroll
        for (int kk = 0; kk < KC; kk += 4) {
            v2f a0 = *(const v2f*)&Xs[wm * 32 + rsel     ][kk + koff];
            v2f a1 = *(const v2f*)&Xs[wm * 32 + rsel + 16][kk + koff];
            v2f b0 = *(const v2f*)&Ws[wn * 32 + rsel     ][kk + koff];
            v2f b1 = *(const v2f*)&Ws[wn * 32 + rsel + 16][kk + koff];
            acc00 = wmma4(a0, b0, acc00);
            acc01 = wmma4(a0, b1, acc01);
            acc10 = wmma4(a1, b0, acc10);
            acc11 = wmma4(a1, b1, acc11);
        }
        __syncthreads();
    }

    const int m0 = mblk + wm * 32;
    const int n0 = nblk + wn * 32;
    store_tile(acc00, m0,      n0,      rsel, hi8, Y, bias, applyTanh);
    store_tile(acc01, m0,      n0 + 16, rsel, hi8, Y, bias, applyTanh);
    store_tile(acc10, m0 + 16, n0,      rsel, hi8, Y, bias, applyTanh);
    store_tile(acc11, m0 + 16, n0 + 16, rsel, hi8, Y, bias, applyTanh);
}

__device__ __forceinline__ float sigmoid_fast(float x) {
    return 1.0f / (1.0f + __expf(-x));
}

// Sequential scan over t. One thread per (b,d) element: 32768 threads =
// 1024 wave32s. outs[t] holds alpha_x on entry (overwritten with out),
// hbuf[t+1] holds v_t on entry (overwritten with h_{t+1}).
__global__ __launch_bounds__(256) void gated_scan_kernel(
    const float* __restrict__ h0,
    const float* __restrict__ d_alpha, const float* __restrict__ b_alpha,
    float* __restrict__ outs, float* __restrict__ hbuf)
{
    const int idx = blockIdx.x * blockDim.x + threadIdx.x;  // 0..BD-1
    const int d   = idx & (DIM - 1);

    float h  = h0[idx];
    const float da = d_alpha[d];
    const float ba = b_alpha[d];

    hbuf[idx] = h;                       // h[0] = h0

    float* op = outs + idx;              // alpha_x[t] / out[t]
    float* hp = hbuf + BD + idx;         // v[t] / h[t+1]

    for (int t = 0; t < T_STEPS; ++t) {
        const float ax = *op;
        const float vv = *hp;
        __builtin_prefetch(op + BD, 0, 1);   // global_prefetch_b8, next t
        __builtin_prefetch(hp + BD, 0, 1);

        const float alpha = sigmoid_fast(ax + da * h + ba);
        h = alpha * h + (1.0f - alpha) * vv;
        const float out = h * h * sigmoid_fast(h);

        *op = out;
        *hp = h;
        op += BD;
        hp += BD;
    }
}

extern "C" void kernel_launch(void* const* d_in, const int* in_sizes, int n_in,
                              void* d_out, int out_size, void* d_ws, size_t ws_size,
                              hipStream_t stream) {
    (void)in_sizes; (void)n_in; (void)out_size; (void)d_ws; (void)ws_size;

    const float* x       = (const float*)d_in[0];  // [T,B,D]
    const float* h0      = (const float*)d_in[1];  // [B,D]
    const float* W_alpha = (const float*)d_in[2];  // [D,D]
    const float* d_alpha = (const float*)d_in[3];  // [D]
    const float* b_alpha = (const float*)d_in[4];  // [D]
    const float* W_x     = (const float*)d_in[5];  // [D,D]
    const float* b_v     = (const float*)d_in[6];  // [D]

    float* outs = (float*)d_out;                   // [T,B,D]
    float* hbuf = outs + (size_t)T_STEPS * BD;     // [T+1,B,D]

    dim3 ggrid(MTOT / BM, DIM / BN);               // 512 x 8 blocks

    // alpha_x -> outs region (raw accumulators; scan applies the sigmoid)
    gemm_xwt_kernel<<<ggrid, 256, 0, stream>>>(x, W_alpha, outs, nullptr, 0);
    // v = tanh(x @ W_x^T + b_v) -> hbuf[1..T] region
    gemm_xwt_kernel<<<ggrid, 256, 0, stream>>>(x, W_x, hbuf + BD, b_v, 1);

    // Sequential gated scan, fully coalesced streaming over d_out
    gated_scan_kernel<<<BD / 256, 256, 0, stream>>>(h0, d_alpha, b_alpha, outs, hbuf);
}